// Encoder_68659347194134
// MI455X (gfx1250) — compile-verified
//
#include <hip/hip_runtime.h>

typedef __attribute__((ext_vector_type(2))) float v2f;
typedef __attribute__((ext_vector_type(8))) float v8f;

// ---------------------------------------------------------------------------
// Gather -> WMMA(f32 16x16x4) -> scatter-atomic sparse conv.
//   X   : (n_in, CIN)  input features (no pad row; guarded gather)
//   W   : (K, CIN, COUT)
//   km_in/km_out : (K, P) pair lists; padding entries hold n_in / n_out
//   acc : (n_out+1, COUT) zero-initialized accumulator (+1 pad row absorbs
//         padded scatters)
// One wave handles a 16-pair tile of one offset k. 4 waves per block.
// ---------------------------------------------------------------------------
template<int CIN, int COUT>
__global__ __launch_bounds__(128) void sparse_conv_wmma_f32(
    const float* __restrict__ X, int n_in,
    const float* __restrict__ W,
    const int* __restrict__ km_in, const int* __restrict__ km_out,
    int P, int n_out, float* __restrict__ acc)
{
    const int k     = blockIdx.y;
    const int wave  = threadIdx.x >> 5;
    const int lane  = threadIdx.x & 31;
    const int tile0 = (blockIdx.x * 4 + wave) * 16;
    if (tile0 >= P) return;            // wave-uniform: EXEC stays all-ones

    const int kbase = k * P;
    const int col   = lane & 15;       // A-row (gather) / B,D column index
    const int half  = lane >> 4;       // selects K pair (A/B) and M half (D)

    // --- gather A fragments (16 x CIN panel), ISA 16x4 f32 A layout ---
    const int  pa = tile0 + col;
    const int  ii = (pa < P) ? km_in[kbase + pa] : n_in;
    const bool av = (ii < n_in);       // n_in == padded "zero row"

    constexpr int KCH = (CIN + 3) / 4;
    v2f afrag[KCH];
#pragma unroll
    for (int kc = 0; kc < KCH; ++kc) {
        const int kk = kc * 4 + 2 * half;
        float ax = (av && kk     < CIN) ? X[ii * CIN + kk    ] : 0.0f;
        float ay = (av && kk + 1 < CIN) ? X[ii * CIN + kk + 1] : 0.0f;
        v2f a; a.x = ax; a.y = ay;
        afrag[kc] = a;
    }

    const float* Wk = W + k * CIN * COUT;
#pragma unroll
    for (int nc = 0; nc < COUT / 16; ++nc) {
        const int ncol = nc * 16 + col;
        v8f c = {};
#pragma unroll
        for (int kc = 0; kc < KCH; ++kc) {
            const int kk = kc * 4 + 2 * half;
            float bx = (kk     < CIN) ? Wk[(kk    ) * COUT + ncol] : 0.0f;
            float by = (kk + 1 < CIN) ? Wk[(kk + 1) * COUT + ncol] : 0.0f;
            v2f b; b.x = bx; b.y = by;
            // D = A(16x4) * B(4x16) + C, full f32
            c = __builtin_amdgcn_wmma_f32_16x16x4_f32(
                    false, afrag[kc], false, b, (short)0, c, false, false);
        }
        // --- scatter D (VGPR v holds row m = v + 8*half, column = ncol) ---
#pragma unroll
        for (int v = 0; v < 8; ++v) {
            const int pm = tile0 + v + 8 * half;
            const int jo = (pm < P) ? km_out[kbase + pm] : n_out;
            __hip_atomic_fetch_add(&acc[jo * COUT + ncol], c[v],
                                   __ATOMIC_RELAXED, __HIP_MEMORY_SCOPE_AGENT);
        }
    }
}

// ---------------------------------------------------------------------------
// Fused bias (+residual) (+ReLU) epilogue; optional second destination
// (used once, to emit the `cached` output alongside the next layer's input).
// ---------------------------------------------------------------------------
template<int C, bool RELU, bool RES>
__global__ void finalize_kernel(const float* __restrict__ acc,
                                const float* __restrict__ bias,
                                const float* __restrict__ res,
                                float* __restrict__ dst,
                                float* __restrict__ dst2,
                                int n_out)
{
    const int idx = blockIdx.x * blockDim.x + threadIdx.x;
    if (idx >= n_out * C) return;
    float v = acc[idx] + bias[idx & (C - 1)];
    if (RES)  v += res[idx];
    if (RELU) v = fmaxf(v, 0.0f);
    dst[idx] = v;
    if (dst2) dst2[idx] = v;
}

static inline int cdiv_i(int a, int b) { return (a + b - 1) / b; }

extern "C" void kernel_launch(void* const* d_in, const int* in_sizes, int n_in_cnt,
                              void* d_out, int out_size, void* d_ws, size_t ws_size,
                              hipStream_t stream)
{
    (void)n_in_cnt; (void)ws_size;

    const float* in_feats = (const float*)d_in[0];
    const float* W_first  = (const float*)d_in[1];
    const float* b_first  = (const float*)d_in[2];
    const float* W_pre    = (const float*)d_in[3];
    const float* b_pre    = (const float*)d_in[4];
    const float* W_down   = (const float*)d_in[5];
    const float* b_down   = (const float*)d_in[6];
    const float* W_r0     = (const float*)d_in[7];
    const float* b_r0     = (const float*)d_in[8];
    const float* W_r1     = (const float*)d_in[9];
    const float* b_r1     = (const float*)d_in[10];
    const float* W_fin    = (const float*)d_in[11];
    const float* b_fin    = (const float*)d_in[12];
    const int*   km0_in   = (const int*)d_in[13];
    const int*   km0_out  = (const int*)d_in[14];
    const int*   kmd_in   = (const int*)d_in[15];
    const int*   kmd_out  = (const int*)d_in[16];
    const int*   km1_in   = (const int*)d_in[17];
    const int*   km1_out  = (const int*)d_in[18];

    const int n0 = in_sizes[0];              // C_IN == 1
    const int P0 = in_sizes[13] / 27;
    const int Pd = in_sizes[15] / 8;
    const int P1 = in_sizes[17] / 27;
    const int n1 = (out_size - n0 * 16) / 32;  // out = n1*32 floats, cached = n0*16

    float* out    = (float*)d_out;              // (n1, 32)
    float* cached = out + (size_t)n1 * 32;      // (n0, 16)

    // ---- workspace layout (floats) ----
    float* ws = (float*)d_ws;
    const size_t acc0N = (size_t)(n0 + 1) * 16;
    const size_t acc1N = (size_t)(n1 + 1) * 32;
    const size_t accN  = (acc0N > acc1N) ? acc0N : acc1N;
    float* acc  = ws;
    float* bufA = acc  + accN;                  // first-block output (n0,16)
    float* bufB = bufA + (size_t)n0 * 16;       // pre-conv output   (n0,16)
    float* bufC = bufB + (size_t)n0 * 16;       // level-1 features  (n1,32)
    float* bufD = bufC + (size_t)n1 * 32;       // resblock temp     (n1,32)

    const dim3 blk(128);

    // ---- L1: first conv 1->16 over km0, bias+relu, dual-write cached ----
    hipMemsetAsync(acc, 0, acc0N * sizeof(float), stream);
    sparse_conv_wmma_f32<1, 16><<<dim3(cdiv_i(P0, 64), 27), blk, 0, stream>>>(
        in_feats, n0, W_first, km0_in, km0_out, P0, n0, acc);
    finalize_kernel<16, true, false><<<cdiv_i(n0 * 16, 256), 256, 0, stream>>>(
        acc, b_first, nullptr, bufA, cached, n0);

    // ---- L2: pre conv 16->16 over km0, bias+relu ----
    hipMemsetAsync(acc, 0, acc0N * sizeof(float), stream);
    sparse_conv_wmma_f32<16, 16><<<dim3(cdiv_i(P0, 64), 27), blk, 0, stream>>>(
        bufA, n0, W_pre, km0_in, km0_out, P0, n0, acc);
    finalize_kernel<16, true, false><<<cdiv_i(n0 * 16, 256), 256, 0, stream>>>(
        acc, b_pre, nullptr, bufB, nullptr, n0);

    // ---- L3: stride-2 down conv 16->32 over kmd (8 offsets), bias+relu ----
    hipMemsetAsync(acc, 0, acc1N * sizeof(float), stream);
    sparse_conv_wmma_f32<16, 32><<<dim3(cdiv_i(Pd, 64), 8), blk, 0, stream>>>(
        bufB, n0, W_down, kmd_in, kmd_out, Pd, n1, acc);
    finalize_kernel<32, true, false><<<cdiv_i(n1 * 32, 256), 256, 0, stream>>>(
        acc, b_down, nullptr, bufC, nullptr, n1);

    // ---- L4: resblock conv0 32->32 over km1, bias+relu ----
    hipMemsetAsync(acc, 0, acc1N * sizeof(float), stream);
    sparse_conv_wmma_f32<32, 32><<<dim3(cdiv_i(P1, 64), 27), blk, 0, stream>>>(
        bufC, n1, W_r0, km1_in, km1_out, P1, n1, acc);
    finalize_kernel<32, true, false><<<cdiv_i(n1 * 32, 256), 256, 0, stream>>>(
        acc, b_r0, nullptr, bufD, nullptr, n1);

    // ---- L5: resblock conv1 32->32, bias + residual(bufC), no relu ----
    hipMemsetAsync(acc, 0, acc1N * sizeof(float), stream);
    sparse_conv_wmma_f32<32, 32><<<dim3(cdiv_i(P1, 64), 27), blk, 0, stream>>>(
        bufD, n1, W_r1, km1_in, km1_out, P1, n1, acc);
    finalize_kernel<32, false, true><<<cdiv_i(n1 * 32, 256), 256, 0, stream>>>(
        acc, b_r1, bufC, bufC, nullptr, n1);   // bufC = r + x (in place)

    // ---- L6: final conv 32->32, bias only, straight into d_out ----
    hipMemsetAsync(acc, 0, acc1N * sizeof(float), stream);
    sparse_conv_wmma_f32<32, 32><<<dim3(cdiv_i(P1, 64), 27), blk, 0, stream>>>(
        bufC, n1, W_fin, km1_in, km1_out, P1, n1, acc);
    finalize_kernel<32, false, false><<<cdiv_i(n1 * 32, 256), 256, 0, stream>>>(
        acc, b_fin, nullptr, out, nullptr, n1);
}